// GCN_49976239456719
// MI455X (gfx1250) — compile-verified
//
#include <hip/hip_runtime.h>
#include <hip/hip_bf16.h>

// ---------------------------------------------------------------------------
// GCN 2-layer forward on MI455X (gfx1250, wave32).
//   layer: H = X*W ; agg = scatter-add over edges with dinv[src]*dinv[dst];
//          out = relu(BN(agg + dinv^2 * H + b))
// GEMMs use V_WMMA_F32_16X16X4_F32 (fp32 matrix pipe, exact fp32 numerics),
// with W staged once per block in LDS (ds_load feeds the WMMA B operand).
// Problem is scatter-bound (~2.5 GB edge traffic vs 5 GFLOP of GEMM).
// ---------------------------------------------------------------------------

typedef float v2f __attribute__((ext_vector_type(2)));
typedef float v8f __attribute__((ext_vector_type(8)));

#define N_NODES 100000
#define N_EDGES 1600000
#define IN_DIM  128
#define HID_DIM 128
#define OUT_DIM 64
#define BN_EPS  1e-5f

// ---------------- utility: zero a region (float4 granularity) -------------
__global__ void zero_f4(float4* __restrict__ p, int n4) {
    int i = blockIdx.x * blockDim.x + threadIdx.x;
    if (i < n4) p[i] = make_float4(0.f, 0.f, 0.f, 0.f);
}

// ---------------- degree count (int hardware atomics) ----------------------
__global__ void deg_count(const long long* __restrict__ ei, unsigned* __restrict__ deg, int E) {
    int e = blockIdx.x * blockDim.x + threadIdx.x;
    if (e < E) {
        long long d = ei[E + e];        // row 1 = dst
        atomicAdd(&deg[d], 1u);
    }
}

__global__ void make_dinv(const unsigned* __restrict__ deg, float* __restrict__ dinv, int n) {
    int i = blockIdx.x * blockDim.x + threadIdx.x;
    if (i < n) dinv[i] = rsqrtf((float)deg[i] + 1.0f);   // deg includes self-loop
}

// ---------------- fp32 WMMA GEMM: H[M x NCOL] = A[M x 128] * W[128 x NCOL] -
// One wave computes a 16-row strip across all NCOL/16 column tiles.
// W is staged in LDS once per block (8 waves share it -> 8x less fetch).
// Fragment layout per CDNA5 ISA 7.12.2 (32-bit A 16x4 / C 16x16 tables):
//   lane = {kh, m}: m = lane&15, kh = lane>>4
//   A/B operand VGPR j  -> K = k0 + j + 2*kh
//   D       VGPR j      -> M = j + 8*kh, N = lane&15
template<int NCOL>
__global__ void gemm_wmma_f32(const float* __restrict__ A,
                              const float* __restrict__ W,
                              float* __restrict__ H, int M) {
    extern __shared__ float wsh[];                    // [128 * NCOL] fp32

    // ---- cooperative stage of W into LDS (all threads; barrier-safe) ----
    {
        constexpr int TOT4 = (128 * NCOL) / 4;
        float4*       ds = (float4*)wsh;
        const float4* gs = (const float4*)W;
        for (int i = threadIdx.x; i < TOT4; i += 256) ds[i] = gs[i];
    }
    __syncthreads();

    const int lane   = threadIdx.x & 31;
    const int waveId = blockIdx.x * (blockDim.x >> 5) + (threadIdx.x >> 5);
    const int row0   = waveId * 16;
    if (row0 < M) {                                   // wave-uniform: EXEC all-1 at WMMA
        const int m  = lane & 15;                     // doubles as N index for B/D
        const int kh = lane >> 4;                     // 0 or 1

        constexpr int NT = NCOL / 16;
        v8f acc[NT];
#pragma unroll
        for (int t = 0; t < NT; ++t) acc[t] = (v8f){0.f,0.f,0.f,0.f,0.f,0.f,0.f,0.f};

        const float* __restrict__ arow = A + (size_t)(row0 + m) * 128;

        for (int k0 = 0; k0 < 128; k0 += 4) {
            const int k = k0 + 2 * kh;
            // A fragment: 16x4 tile, 2 consecutive fp32 per lane -> one b64 load
            const float2 av = *(const float2*)(arow + k);
            v2f a; a.x = av.x; a.y = av.y;
            const float* __restrict__ wrow = wsh + k * NCOL + m;
#pragma unroll
            for (int t = 0; t < NT; ++t) {
                // B fragment: 4x16 tile of W from LDS (ds_load)
                v2f b; b.x = wrow[t * 16]; b.y = wrow[t * 16 + NCOL];
                acc[t] = __builtin_amdgcn_wmma_f32_16x16x4_f32(
                    false, a, false, b, (short)0, acc[t], false, false);
            }
        }

#pragma unroll
        for (int t = 0; t < NT; ++t) {
#pragma unroll
            for (int j = 0; j < 8; ++j) {
                const int mm = j + 8 * kh;
                H[(size_t)(row0 + mm) * NCOL + t * 16 + m] = acc[t][j];
            }
        }
    }
}

// ---------------- edge scatter: agg[dst] += h[src] * dinv[src]*dinv[dst] ---
// One wave per edge; lane handles F/32 consecutive channels.
// Gather is one b128/b64 load per lane; scatter uses fp32 HW atomics.
template<int F>
__global__ void scatter_edges(const float* __restrict__ h,
                              const long long* __restrict__ ei,
                              const float* __restrict__ dinv,
                              float* __restrict__ agg, int E) {
    const int lane = threadIdx.x & 31;
    const int e    = (blockIdx.x * blockDim.x + threadIdx.x) >> 5;
    if (e >= E) return;
    const long long s = ei[e];
    const long long d = ei[E + e];
    const float w = dinv[s] * dinv[d];

    constexpr int C = F / 32;                       // 4 (F=128) or 2 (F=64)
    const float* __restrict__ hp = h + (size_t)s * F + lane * C;
    float* __restrict__ ap = agg + (size_t)d * F + lane * C;
    __builtin_prefetch(ap, 1, 0);                   // global_prefetch_b8 on gfx1250
    if constexpr (C == 4) {
        const float4 hv = *(const float4*)hp;       // global_load_b128
        unsafeAtomicAdd(ap + 0, hv.x * w);          // global_atomic_add_f32
        unsafeAtomicAdd(ap + 1, hv.y * w);
        unsafeAtomicAdd(ap + 2, hv.z * w);
        unsafeAtomicAdd(ap + 3, hv.w * w);
    } else {
        const float2 hv = *(const float2*)hp;       // global_load_b64
        unsafeAtomicAdd(ap + 0, hv.x * w);
        unsafeAtomicAdd(ap + 1, hv.y * w);
    }
}

// ---------------- fused self-loop + bias + BN(eval) + ReLU -----------------
template<int F>
__global__ void post_bn_relu(const float* __restrict__ agg,
                             const float* __restrict__ h,
                             const float* __restrict__ dinv,
                             const float* __restrict__ bias,
                             const float* __restrict__ g,
                             const float* __restrict__ be,
                             const float* __restrict__ rm,
                             const float* __restrict__ rv,
                             float* __restrict__ out, int n) {
    int i = blockIdx.x * blockDim.x + threadIdx.x;
    if (i >= n * F) return;
    const int node = i / F;
    const int c    = i - node * F;
    const float di = dinv[node];
    float v = agg[i] + h[i] * (di * di) + bias[c];
    v = (v - rm[c]) * rsqrtf(rv[c] + BN_EPS) * g[c] + be[c];
    out[i] = fmaxf(v, 0.f);
}

// ---------------------------------------------------------------------------
extern "C" void kernel_launch(void* const* d_in, const int* in_sizes, int n_in,
                              void* d_out, int out_size, void* d_ws, size_t ws_size,
                              hipStream_t stream) {
    const float*      x  = (const float*)d_in[0];
    const long long*  ei = (const long long*)d_in[1];   // int64 [2, E]
    const float* W1  = (const float*)d_in[2];
    const float* b1  = (const float*)d_in[3];
    const float* g1  = (const float*)d_in[4];
    const float* be1 = (const float*)d_in[5];
    const float* rm1 = (const float*)d_in[6];
    const float* rv1 = (const float*)d_in[7];
    const float* W2  = (const float*)d_in[8];
    const float* b2  = (const float*)d_in[9];
    const float* g2  = (const float*)d_in[10];
    const float* be2 = (const float*)d_in[11];
    const float* rm2 = (const float*)d_in[12];
    const float* rv2 = (const float*)d_in[13];
    float* out = (float*)d_out;

    // Workspace layout (bytes): ~98.4 MB total
    char* ws = (char*)d_ws;
    const size_t H1_BYTES  = (size_t)N_NODES * HID_DIM * sizeof(float);   // 51.2 MB
    const size_t B_BYTES   = H1_BYTES;                                    // 51.2 MB (agg1; later agg2+h2)
    float*    h1   = (float*)(ws);                       // X*W1, then h1act (in place)
    float*    agg1 = (float*)(ws + H1_BYTES);            // layer-1 aggregation
    float*    agg2 = (float*)(ws + H1_BYTES);            // reuse: first 25.6 MB
    float*    h2   = (float*)(ws + H1_BYTES + (size_t)N_NODES * OUT_DIM * sizeof(float));
    unsigned* deg  = (unsigned*)(ws + H1_BYTES + B_BYTES);
    float*    dinv = (float*)(ws + H1_BYTES + B_BYTES + (size_t)N_NODES * sizeof(unsigned));

    const int TB = 256;

    // 1) zero agg1 + deg
    {
        int n4 = (N_NODES * HID_DIM) / 4;
        zero_f4<<<(n4 + TB - 1) / TB, TB, 0, stream>>>((float4*)agg1, n4);
        int d4 = N_NODES / 4;
        zero_f4<<<(d4 + TB - 1) / TB, TB, 0, stream>>>((float4*)deg, d4);
    }
    // 2) degree + dinv
    deg_count<<<(N_EDGES + TB - 1) / TB, TB, 0, stream>>>(ei, deg, N_EDGES);
    make_dinv<<<(N_NODES + TB - 1) / TB, TB, 0, stream>>>(deg, dinv, N_NODES);

    // 3) GEMM1: h1 = x @ W1  (6250 waves, 8 waves/block, 64KB LDS for W1)
    {
        int waves  = N_NODES / 16;
        int blocks = (waves + 7) / 8;
        gemm_wmma_f32<HID_DIM><<<blocks, TB, 128 * HID_DIM * sizeof(float), stream>>>(
            x, W1, h1, N_NODES);
    }
    // 4) scatter layer 1
    {
        long long thr = (long long)N_EDGES * 32;
        scatter_edges<HID_DIM><<<(int)((thr + TB - 1) / TB), TB, 0, stream>>>(h1, ei, dinv, agg1, N_EDGES);
    }
    // 5) epilogue layer 1 (in place into h1 -> h1act)
    post_bn_relu<HID_DIM><<<(N_NODES * HID_DIM + TB - 1) / TB, TB, 0, stream>>>(
        agg1, h1, dinv, b1, g1, be1, rm1, rv1, h1, N_NODES);

    // 6) zero agg2
    {
        int n4 = (N_NODES * OUT_DIM) / 4;
        zero_f4<<<(n4 + TB - 1) / TB, TB, 0, stream>>>((float4*)agg2, n4);
    }
    // 7) GEMM2: h2 = h1act @ W2  (32KB LDS for W2)
    {
        int waves  = N_NODES / 16;
        int blocks = (waves + 7) / 8;
        gemm_wmma_f32<OUT_DIM><<<blocks, TB, 128 * OUT_DIM * sizeof(float), stream>>>(
            h1, W2, h2, N_NODES);
    }
    // 8) scatter layer 2
    {
        long long thr = (long long)N_EDGES * 32;
        scatter_edges<OUT_DIM><<<(int)((thr + TB - 1) / TB), TB, 0, stream>>>(h2, ei, dinv, agg2, N_EDGES);
    }
    // 9) epilogue layer 2 -> d_out
    post_bn_relu<OUT_DIM><<<(N_NODES * OUT_DIM + TB - 1) / TB, TB, 0, stream>>>(
        agg2, h2, dinv, b2, g2, be2, rm2, rv2, out, N_NODES);
}